// Router_27195732918428
// MI455X (gfx1250) — compile-verified
//
#include <hip/hip_runtime.h>
#include <math.h>
#include <stdint.h>

typedef __attribute__((ext_vector_type(2))) float v2f;
typedef __attribute__((ext_vector_type(8))) float v8f;
typedef __attribute__((ext_vector_type(4))) unsigned int u32x4;
typedef __attribute__((ext_vector_type(8))) int i32x8;
typedef __attribute__((ext_vector_type(4))) int i32x4;

#define DDIM 2048
#define NEXP 64
#define TOKB 128                 // tokens per block (8 waves x 16)
#define KC 128                   // K-chunk staged per TDM round
#define NCHUNK (DDIM / KC)       // 16
#define LSTR 130                 // padded LDS row stride in floats (128 + 2, even for b64 loads)
#define XBYTES (TOKB * LSTR * 4) // 66560
#define WBYTES (NEXP * LSTR * 4) // 33280
#define BUFBYTES (XBYTES + WBYTES)
#define SMEM_BYTES (2 * BUFBYTES) // 199680 dynamic LDS
#define LOG_STR 65               // padded logits stride

extern __shared__ char smem[];

__device__ __forceinline__ unsigned bufX(int p) { return (unsigned)(p * BUFBYTES); }
__device__ __forceinline__ unsigned bufW(int p) { return (unsigned)(p * BUFBYTES + XBYTES); }

// Issue one TDM 2D tile load: tile_rows x KC f32 elements, row stride DDIM in
// global memory, into LDS at lds_off with 2-DWORD padding every 128 DWORDs
// (-> LSTR-float rows). Descriptor layout per CDNA5 ISA.
__device__ __forceinline__ void tdm_load_2d(unsigned lds_off, const float* gsrc,
                                            unsigned tile_rows, unsigned tensor_rows) {
  uint64_t ga = (uint64_t)(uintptr_t)gsrc;
  u32x4 g0;
  g0[0] = 1u;                                            // count=1, user desc, gather off
  g0[1] = lds_off;                                       // lds_addr (bytes)
  g0[2] = (unsigned)ga;                                  // global_addr[31:0]
  g0[3] = (unsigned)((ga >> 32) & 0x01FFFFFFu) | 0x80000000u; // global_addr[56:32] | type=2
  i32x8 g1;
  g1[0] = (2 << 16)        // data_size = 4B
        | (1 << 20)        // pad_enable
        | (6 << 22)        // pad_interval: every 128 DWORDs
        | (1 << 25);       // pad_amount: 2 DWORDs
  g1[1] = (int)((unsigned)DDIM << 16);                   // tensor_dim0 lo16 (=2048)
  g1[2] = (int)((tensor_rows & 0xFFFFu) << 16);          // tensor_dim0 hi16=0 | tensor_dim1 lo16
  g1[3] = (int)(((tensor_rows >> 16) & 0xFFFFu) | ((unsigned)KC << 16)); // dim1 hi16 | tile_dim0
  g1[4] = (int)(tile_rows & 0xFFFFu);                    // tile_dim1 | tile_dim2=0
  g1[5] = DDIM;                                          // tensor_dim0_stride lo32
  g1[6] = 0;                                             // stride0 hi16 | stride1 lo16
  g1[7] = 0;                                             // stride1 hi32
  i32x4 z4 = {0, 0, 0, 0};
  i32x8 z8 = {0, 0, 0, 0, 0, 0, 0, 0};
  __builtin_amdgcn_tensor_load_to_lds(g0, g1, z4, z4, z8, 0);
}

__global__ __launch_bounds__(256)
void router_wmma_kernel(const float* __restrict__ x,
                        const float* __restrict__ W,
                        float* __restrict__ out,
                        int tokens) {
  const int lane = threadIdx.x & 31;
  const int wave = threadIdx.x >> 5;
  const int m    = lane & 15;         // row within 16-tile
  const int kb   = (lane >> 4) << 1;  // K sub-offset 0 or 2 (f32 16x16x4 A/B layout)
  const int tokenBase = blockIdx.x * TOKB;

  v8f c0 = {}, c1 = {}, c2 = {}, c3 = {};

  // --- Prologue: stage chunk 0 ---
  if (wave == 0) {
    tdm_load_2d(bufX(0), x + (size_t)tokenBase * DDIM, TOKB, (unsigned)tokens);
    tdm_load_2d(bufW(0), W, NEXP, NEXP);
  }

  for (int c = 0; c < NCHUNK; ++c) {
    if (wave == 0) {
      if (c + 1 < NCHUNK) {
        const int k = (c + 1) * KC;
        tdm_load_2d(bufX((c + 1) & 1), x + (size_t)tokenBase * DDIM + k, TOKB, (unsigned)tokens);
        tdm_load_2d(bufW((c + 1) & 1), W + k, NEXP, NEXP);
        __builtin_amdgcn_s_wait_tensorcnt(2);  // chunk c's 2 TDM ops done (in-order)
      } else {
        __builtin_amdgcn_s_wait_tensorcnt(0);
      }
    }
    __syncthreads();  // publish chunk c to all waves

    const float* lx = (const float*)(smem + bufX(c & 1)) + (wave * 16 + m) * LSTR + kb;
    const float* lw = (const float*)(smem + bufW(c & 1)) + m * LSTR + kb;
#pragma unroll 4
    for (int kk = 0; kk < KC; kk += 4) {
      v2f a  = *(const v2f*)(lx + kk);
      v2f b0 = *(const v2f*)(lw + kk);
      v2f b1 = *(const v2f*)(lw + 16 * LSTR + kk);
      v2f b2 = *(const v2f*)(lw + 32 * LSTR + kk);
      v2f b3 = *(const v2f*)(lw + 48 * LSTR + kk);
      c0 = __builtin_amdgcn_wmma_f32_16x16x4_f32(false, a, false, b0, (short)0, c0, false, false);
      c1 = __builtin_amdgcn_wmma_f32_16x16x4_f32(false, a, false, b1, (short)0, c1, false, false);
      c2 = __builtin_amdgcn_wmma_f32_16x16x4_f32(false, a, false, b2, (short)0, c2, false, false);
      c3 = __builtin_amdgcn_wmma_f32_16x16x4_f32(false, a, false, b3, (short)0, c3, false, false);
    }
    __syncthreads();  // all waves done reading buf (c&1) before it is re-staged
  }

  // --- Logits to LDS (overlays staging buffers; accumulators hold all state) ---
  float* L      = (float*)smem;                               // TOKB x LOG_STR
  float* s_max  = (float*)(smem + TOKB * LOG_STR * 4);
  float* s_inv  = s_max + TOKB;
  float* s_rpn  = s_inv + TOKB;
  int*   s_i0   = (int*)(s_rpn + TOKB);
  int*   s_i1   = s_i0 + TOKB;

  {
    const int tRow = wave * 16 + ((lane >> 4) << 3);  // C/D layout: VGPR j -> M=j (+8 hi half)
    const int e = lane & 15;
#pragma unroll
    for (int j = 0; j < 8; ++j) {
      float* row = &L[(tRow + j) * LOG_STR + e];
      row[0]  = c0[j];
      row[16] = c1[j];
      row[32] = c2[j];
      row[48] = c3[j];
    }
  }
  __syncthreads();

  // Output layout (flat, return order): mask [T,64] | idx [T,2] | router_probs [T,64] | probs [T,64]
  const int T = tokens;
  float* out_mask = out;
  float* out_idx  = out + (size_t)T * NEXP;
  float* out_rp   = out + (size_t)T * (NEXP + 2);
  float* out_p    = out + (size_t)T * (2 * NEXP + 2);

  if (threadIdx.x < TOKB) {
    const int t = threadIdx.x;
    const float* row = &L[t * LOG_STR];
    float m1 = row[0]; int i0 = 0;
    float m2 = -INFINITY; int i1 = -1;
#pragma unroll
    for (int e = 1; e < NEXP; ++e) {
      float v = row[e];
      if (v > m1)      { m2 = m1; i1 = i0; m1 = v; i0 = e; }
      else if (v > m2) { m2 = v; i1 = e; }
    }
    float s = 0.f;
#pragma unroll
    for (int e = 0; e < NEXP; ++e) s += __expf(row[e] - m1);
    const float inv = 1.f / s;
    const float p0 = inv;
    const float p1 = __expf(m2 - m1) * inv;
    s_max[t] = m1;
    s_inv[t] = inv;
    s_i0[t]  = i0;
    s_i1[t]  = i1;
    s_rpn[t] = 1.f / (p0 + p1);
    const size_t g = (size_t)(tokenBase + t) * 2;
    out_idx[g + 0] = (float)i0;
    out_idx[g + 1] = (float)i1;
  }
  __syncthreads();

  for (int it = 0; it < (TOKB * NEXP) / 256; ++it) {
    const int idx = it * 256 + threadIdx.x;
    const int t = idx >> 6;
    const int e = idx & (NEXP - 1);
    const float v = L[t * LOG_STR + e];
    const float p = __expf(v - s_max[t]) * s_inv[t];
    const bool top = (e == s_i0[t]) || (e == s_i1[t]);
    const size_t g = (size_t)(tokenBase + t) * NEXP + e;
    out_mask[g] = top ? 1.f : 0.f;
    out_rp[g]   = top ? p * s_rpn[t] : 0.f;
    out_p[g]    = p;
  }
}

extern "C" void kernel_launch(void* const* d_in, const int* in_sizes, int n_in,
                              void* d_out, int out_size, void* d_ws, size_t ws_size,
                              hipStream_t stream) {
  const float* x = (const float*)d_in[0];
  const float* W = (const float*)d_in[1];
  float* out = (float*)d_out;
  const int tokens = in_sizes[0] / DDIM;      // 16384
  const int blocks = tokens / TOKB;           // 128
  (void)hipFuncSetAttribute((const void*)router_wmma_kernel,
                            hipFuncAttributeMaxDynamicSharedMemorySize, SMEM_BYTES);
  router_wmma_kernel<<<blocks, 256, SMEM_BYTES, stream>>>(x, W, out, tokens);
}